// DJModel_26611617366458
// MI455X (gfx1250) — compile-verified
//
#include <hip/hip_runtime.h>
#include <stdint.h>

// Problem dimensions (fixed by the reference).
#define M_DIM 4096
#define TX_DIM 128
#define NV_DIM 256   // n_values (input / output feature size)
#define NA_DIM 512   // n_a (hidden size)

// GEMM blocking.
#define BM 128            // batch rows per workgroup
#define BN 128            // gate-cols per workgroup (cell: 4 gates x 32 hidden cols)
#define KC 64             // K chunk staged in LDS
#define LDA 68            // LDS row stride (KC + 4 floats padding -> bank spread)
#define THREADS 256       // 8 waves

typedef __attribute__((ext_vector_type(2))) float v2f;
typedef __attribute__((ext_vector_type(8))) float v8f;

__device__ __forceinline__ float fast_sigmoid(float x) {
    return 1.0f / (1.0f + __expf(-x));
}
__device__ __forceinline__ float fast_tanh(float x) {
    float e = __expf(2.0f * x);
    return (e - 1.0f) / (e + 1.0f);
}

// D = A(16x4 f32) x B(4x16 f32) + C, wave32 WMMA.
__device__ __forceinline__ v8f wmma4(v2f a, v2f b, v8f c) {
    return __builtin_amdgcn_wmma_f32_16x16x4_f32(
        /*neg_a=*/false, a, /*neg_b=*/false, b,
        /*c_mod=*/(short)0, c, /*reuse_a=*/false, /*reuse_b=*/false);
}

// Async DMA of 16 bytes global -> LDS (ASYNCcnt-tracked, no VGPR data path).
__device__ __forceinline__ void async_cp16(const float* g, const float* lds) {
    asm volatile("global_load_async_to_lds_b128 %0, %1, off"
                 :: "v"((unsigned)(uintptr_t)lds), "v"(g)
                 : "memory");
}

// Issue one K-chunk's async copies (16 x b128 per thread: 8 for A, 8 for B).
// Per-thread pattern: rows {tid>>4 + 16*i}, 16B column chunk (tid&15)*4.
// B LDS slot row j -> global weight row:
//   GATED: ((j>>5)<<9) + nbase + (j&31)   (4 gates x 32 hidden cols, gate stride 512)
//   else : nbase + j                       (contiguous output cols)
template <bool GATED>
__device__ __forceinline__ void issue_chunk(
    float* __restrict__ Abuf, float* __restrict__ Bbuf, int kb,
    const float* __restrict__ Ag, int strideA, int mbase,
    const float* __restrict__ Bg, int strideB, int nbase, int tid)
{
    const int rbase = tid >> 4;
    const int c = (tid & 15) * 4;
#pragma unroll
    for (int i = 0; i < 8; ++i) {
        const int r = rbase + i * 16;
        async_cp16(Ag + (size_t)(mbase + r) * strideA + kb + c,
                   Abuf + r * LDA + c);
        const int br = GATED ? (((r >> 5) << 9) + nbase + (r & 31))
                             : (nbase + r);
        async_cp16(Bg + (size_t)br * strideB + kb + c,
                   Bbuf + r * LDA + c);
    }
}

// Double-buffered blocked GEMM phase: acc[8] += A[BM x Ktot] * B^T(tile).
// Chunk c+1 streams into the idle LDS buffer while chunk c is multiplied.
// Async loads complete in order, so s_wait_asynccnt 16 (the just-issued next
// chunk) guarantees the current chunk has landed; the barrier publishes it
// across all 8 waves.
template <bool GATED>
__device__ __forceinline__ void gemm_phase(
    float* A0, float* B0, float* A1, float* B1,
    const float* __restrict__ Ag, int strideA, int mbase,
    const float* __restrict__ Bg, int strideB, int nbase,
    int Ktot, v8f acc[8], int tid, int w, int lid, int half)
{
    const int nch = Ktot / KC;
    issue_chunk<GATED>(A0, B0, 0, Ag, strideA, mbase, Bg, strideB, nbase, tid);
    for (int ci = 0; ci < nch; ++ci) {
        float* Ab = (ci & 1) ? A1 : A0;
        float* Bb = (ci & 1) ? B1 : B0;
        if (ci + 1 < nch) {
            float* An = (ci & 1) ? A0 : A1;
            float* Bn = (ci & 1) ? B0 : B1;
            issue_chunk<GATED>(An, Bn, (ci + 1) * KC,
                               Ag, strideA, mbase, Bg, strideB, nbase, tid);
            asm volatile("s_wait_asynccnt 0x10" ::: "memory");
        } else {
            asm volatile("s_wait_asynccnt 0x0" ::: "memory");
        }
        __syncthreads();  // chunk ci visible to all waves
        const float* arow = Ab + (w * 16 + lid) * LDA + half * 2;
#pragma unroll
        for (int kt = 0; kt < KC; kt += 4) {
            v2f a = *(const v2f*)(arow + kt);
#pragma unroll
            for (int j = 0; j < 8; ++j) {
                v2f b = *(const v2f*)(Bb + (j * 16 + lid) * LDA + kt + half * 2);
                acc[j] = wmma4(a, b, acc[j]);
            }
        }
        __syncthreads();  // all waves done reading before buffer is reused
    }
}

// One LSTM timestep (cell WGs) fused with the projection of the PREVIOUS
// step's hidden state (proj WGs). 256 threads / 8 waves per WG.
__global__ __launch_bounds__(THREADS) void lstm_step_kernel(
    const float* __restrict__ Xt,     // element (m,k) at Xt[m*TX*NV + k]
    const float* __restrict__ h_in,   // [M, NA]  (h_{t-1})
    float* __restrict__ h_out,        // [M, NA]  (h_t)
    float* __restrict__ c_st,         // [M, NA]  in-place cell state
    const float* __restrict__ W_ih,   // [4*NA, NV]
    const float* __restrict__ W_hh,   // [4*NA, NA]
    const float* __restrict__ b_ih,   // [4*NA]
    const float* __restrict__ b_hh,   // [4*NA]
    const float* __restrict__ W_out,  // [NV, NA]
    const float* __restrict__ b_out,  // [NV]
    float* __restrict__ out_t,        // [M, NV]  (out for step t-1)
    int cell_wgs)
{
    __shared__ float smem[4 * BM * LDA];   // A0, B0, A1, B1 (double-buffered)
    float* A0 = smem;
    float* B0 = smem + 1 * BM * LDA;
    float* A1 = smem + 2 * BM * LDA;
    float* B1 = smem + 3 * BM * LDA;

    const int tid  = threadIdx.x;
    const int w    = tid >> 5;       // wave id 0..7
    const int lane = tid & 31;
    const int lid  = lane & 15;
    const int half = lane >> 4;
    const int bid  = blockIdx.x;

    v8f acc[8];
#pragma unroll
    for (int j = 0; j < 8; ++j) acc[j] = (v8f){};

    if (bid < cell_wgs) {
        // ---- LSTM cell: 128 rows x (4 gates x 32 hidden cols) ----
        const int n0 = (bid & (NA_DIM / 32 - 1)) << 5;   // hidden col base
        const int m0 = (bid >> 4) << 7;                  // 16 col-groups
        // x_t @ W_ih^T  (K = NV), then h_{t-1} @ W_hh^T (K = NA)
        gemm_phase<true>(A0, B0, A1, B1, Xt, TX_DIM * NV_DIM, m0,
                         W_ih, NV_DIM, n0, NV_DIM, acc, tid, w, lid, half);
        gemm_phase<true>(A0, B0, A1, B1, h_in, NA_DIM, m0,
                         W_hh, NA_DIM, n0, NA_DIM, acc, tid, w, lid, half);

        // Tiles: j = gate*2 + colsub. Pointwise LSTM update, all in-lane.
#pragma unroll
        for (int cs = 0; cs < 2; ++cs) {
            const int n = n0 + cs * 16 + lid;
            const float biasI = b_ih[0 * NA_DIM + n] + b_hh[0 * NA_DIM + n];
            const float biasF = b_ih[1 * NA_DIM + n] + b_hh[1 * NA_DIM + n];
            const float biasG = b_ih[2 * NA_DIM + n] + b_hh[2 * NA_DIM + n];
            const float biasO = b_ih[3 * NA_DIM + n] + b_hh[3 * NA_DIM + n];
            const v8f ai = acc[0 + cs];
            const v8f af = acc[2 + cs];
            const v8f ag = acc[4 + cs];
            const v8f ao = acc[6 + cs];
#pragma unroll
            for (int v = 0; v < 8; ++v) {
                const int m = m0 + w * 16 + half * 8 + v;
                const size_t idx = (size_t)m * NA_DIM + n;
                float c_old = c_st[idx];
                float ig = fast_sigmoid(ai[v] + biasI);
                float fg = fast_sigmoid(af[v] + biasF);
                float gg = fast_tanh(ag[v] + biasG);
                float og = fast_sigmoid(ao[v] + biasO);
                float cn = fg * c_old + ig * gg;
                float hn = og * fast_tanh(cn);
                c_st[idx]  = cn;
                h_out[idx] = hn;
            }
        }
    } else {
        // ---- Output projection of previous step: h_in @ W_out^T + b_out ----
        const int pid = bid - cell_wgs;
        const int n0 = (pid & 1) << 7;                   // 2 col-groups of 128
        const int m0 = (pid >> 1) << 7;
        gemm_phase<false>(A0, B0, A1, B1, h_in, NA_DIM, m0,
                          W_out, NA_DIM, n0, NA_DIM, acc, tid, w, lid, half);
#pragma unroll
        for (int j = 0; j < 8; ++j) {
            const int n = n0 + j * 16 + lid;
            const float bias = b_out[n];
#pragma unroll
            for (int v = 0; v < 8; ++v) {
                const int m = m0 + w * 16 + half * 8 + v;
                out_t[(size_t)m * NV_DIM + n] = acc[j][v] + bias;
            }
        }
    }
}

extern "C" void kernel_launch(void* const* d_in, const int* in_sizes, int n_in,
                              void* d_out, int out_size, void* d_ws, size_t ws_size,
                              hipStream_t stream) {
    (void)in_sizes; (void)n_in; (void)out_size; (void)ws_size;

    const float* X     = (const float*)d_in[0];
    const float* a0    = (const float*)d_in[1];
    const float* c0    = (const float*)d_in[2];
    const float* W_ih  = (const float*)d_in[3];
    const float* W_hh  = (const float*)d_in[4];
    const float* b_ih  = (const float*)d_in[5];
    const float* b_hh  = (const float*)d_in[6];
    const float* W_out = (const float*)d_in[7];
    const float* b_out = (const float*)d_in[8];
    float* out = (float*)d_out;

    // Workspace: h ping, h pong, c (each M*NA f32 = 8 MB).
    float* h0   = (float*)d_ws;
    float* h1   = h0 + (size_t)M_DIM * NA_DIM;
    float* cbuf = h1 + (size_t)M_DIM * NA_DIM;

    const size_t state_bytes = (size_t)M_DIM * NA_DIM * sizeof(float);
    hipMemcpyAsync(h0,   a0, state_bytes, hipMemcpyDeviceToDevice, stream);
    hipMemcpyAsync(cbuf, c0, state_bytes, hipMemcpyDeviceToDevice, stream);

    const int CELL_WG = (M_DIM / BM) * (NA_DIM / 32);   // 32 * 16 = 512
    const int PROJ_WG = (M_DIM / BM) * (NV_DIM / BN);   // 32 * 2  = 64

    float* hin = h0;
    float* hout = h1;
    for (int t = 0; t < TX_DIM; ++t) {
        const int pb = (t > 0) ? PROJ_WG : 0;
        float* out_t = out + (size_t)((t > 0) ? (t - 1) : 0) * M_DIM * NV_DIM;
        lstm_step_kernel<<<dim3(CELL_WG + pb), dim3(THREADS), 0, stream>>>(
            X + (size_t)t * NV_DIM, hin, hout, cbuf,
            W_ih, W_hh, b_ih, b_hh, W_out, b_out, out_t, CELL_WG);
        float* tmp = hin; hin = hout; hout = tmp;
    }
    // Projection of the final hidden state h_{TX-1}.
    lstm_step_kernel<<<dim3(PROJ_WG), dim3(THREADS), 0, stream>>>(
        X, hin, hout, cbuf, W_ih, W_hh, b_ih, b_hh, W_out, b_out,
        out + (size_t)(TX_DIM - 1) * M_DIM * NV_DIM, 0);
}